// ControlModule_11501922419460
// MI455X (gfx1250) — compile-verified
//
#include <hip/hip_runtime.h>

typedef float v2f __attribute__((ext_vector_type(2)));
typedef float v8f __attribute__((ext_vector_type(8)));

#define MT 256            // rows per block: covers worst-case per-expert count => weights stream once
#define NT 64             // output columns per block
#define KT 32             // K chunk
#define LDA 36            // padded LDS row stride (floats): conflict-free, 16B-aligned rows

__global__ void zero_f32(float* __restrict__ p, int n) {
  int i = blockIdx.x * blockDim.x + threadIdx.x;
  if (i < n) p[i] = 0.0f;
}

__global__ void zero_i32(int* __restrict__ p, int n) {
  int i = blockIdx.x * blockDim.x + threadIdx.x;
  if (i < n) p[i] = 0;
}

__global__ void route_kernel(const int* __restrict__ idx, int* __restrict__ counts,
                             int* __restrict__ lists, int B, int E) {
  int i = blockIdx.x * blockDim.x + threadIdx.x;
  if (i < B) {
    int e = idx[i];
    if (e >= 0 && e < E) {
      int p = atomicAdd(&counts[e], 1);
      lists[e * B + p] = i;
    }
  }
}

// flat LDS pointer low 32 bits == wave-relative LDS byte offset (LDS aperture keeps offset in addr[31:0])
__device__ __forceinline__ unsigned lds_u32(const void* p) {
  return (unsigned)(unsigned long long)p;
}

// async global->LDS, 16B per lane, GVS mode (SGPR base + per-lane 32-bit byte offset)
__device__ __forceinline__ void async_ld(unsigned ldsdst, unsigned goff, unsigned long long base) {
  asm volatile("global_load_async_to_lds_b128 %0, %1, %2"
               :: "v"(ldsdst), "v"(goff), "s"(base) : "memory");
}
__device__ __forceinline__ void async_ld_nt(unsigned ldsdst, unsigned goff, unsigned long long base) {
  asm volatile("global_load_async_to_lds_b128 %0, %1, %2 th:TH_LOAD_NT"
               :: "v"(ldsdst), "v"(goff), "s"(base) : "memory");
}

#define WMMA_F32(A, Bf, C) \
  __builtin_amdgcn_wmma_f32_16x16x4_f32(false, (A), false, (Bf), (short)0, (C), false, false)

template <bool BOTH>
__device__ __forceinline__ void compute_chunk(const float* __restrict__ as,
                                              const float* __restrict__ bs,
                                              int m0, int mlane, int ksel,
                                              v8f acc0[4], v8f acc1[4]) {
  const float* af0 = &as[(m0 + mlane) * LDA + ksel];
  const float* af1 = af0 + 16 * LDA;
  const float* bf0 = &bs[( 0 + mlane) * LDA + ksel];
  const float* bf1 = &bs[(16 + mlane) * LDA + ksel];
  const float* bf2 = &bs[(32 + mlane) * LDA + ksel];
  const float* bf3 = &bs[(48 + mlane) * LDA + ksel];
#pragma unroll
  for (int kk = 0; kk < KT; kk += 4) {
    v2f b0 = *(const v2f*)(bf0 + kk);
    v2f b1 = *(const v2f*)(bf1 + kk);
    v2f b2 = *(const v2f*)(bf2 + kk);
    v2f b3 = *(const v2f*)(bf3 + kk);
    v2f a0 = *(const v2f*)(af0 + kk);
    acc0[0] = WMMA_F32(a0, b0, acc0[0]);
    acc0[1] = WMMA_F32(a0, b1, acc0[1]);
    acc0[2] = WMMA_F32(a0, b2, acc0[2]);
    acc0[3] = WMMA_F32(a0, b3, acc0[3]);
    if (BOTH) {
      v2f a1 = *(const v2f*)(af1 + kk);
      acc1[0] = WMMA_F32(a1, b0, acc1[0]);
      acc1[1] = WMMA_F32(a1, b1, acc1[1]);
      acc1[2] = WMMA_F32(a1, b2, acc1[2]);
      acc1[3] = WMMA_F32(a1, b3, acc1[3]);
    }
  }
}

__launch_bounds__(256)
__global__ void moe_wmma_kernel(const float* __restrict__ x,
                                const float* __restrict__ W,
                                const int* __restrict__ counts,
                                const int* __restrict__ lists,
                                float* __restrict__ out,
                                int B, int D) {
  __shared__ float As[2][MT * LDA];   // x tile:  256 x KT
  __shared__ float Bs[2][NT * LDA];   // W tile:   64 x KT (row = output col, K contiguous)

  const int e    = blockIdx.z;
  const int mt   = blockIdx.y;
  const int nt   = blockIdx.x;
  const int cnt  = counts[e];
  const int row0 = mt * MT;
  if (row0 >= cnt) return;            // block-uniform early out

  const float* __restrict__ We = W + (size_t)e * D * D + (size_t)(nt * NT) * D;
  const unsigned long long xB  = (unsigned long long)x;
  const unsigned long long wB  = (unsigned long long)We;
  const int t    = threadIdx.x;
  const int lane = t & 31;
  const int wave = t >> 5;

  // ---- async loader mapping: 256 threads x b128 cover 256x32 (A, 8 ea) and 64x32 (B, 2 ea) ----
  const int lk = (t & 7) * 4;         // K offset within chunk (floats)
  const int lr = t >> 3;              // 0..31 base row

  unsigned aoff[8], adst[8];          // global byte offset (chunk 0) / LDS row-part byte offset
#pragma unroll
  for (int r = 0; r < 8; ++r) {
    int gr  = row0 + lr + 32 * r;
    int rid = (gr < cnt) ? lists[e * B + gr] : 0;   // padded rows load row 0 (harmless, never stored)
    aoff[r] = (unsigned)((rid * D + lk) * 4);
    adst[r] = (unsigned)(((lr + 32 * r) * LDA + lk) * 4);
  }
  unsigned boff[2], bdst[2];
#pragma unroll
  for (int r = 0; r < 2; ++r) {
    boff[r] = (unsigned)(((lr + 32 * r) * D + lk) * 4);
    bdst[r] = (unsigned)(((lr + 32 * r) * LDA + lk) * 4);
  }
  const unsigned aBase[2] = { lds_u32(&As[0][0]), lds_u32(&As[1][0]) };
  const unsigned bBase[2] = { lds_u32(&Bs[0][0]), lds_u32(&Bs[1][0]) };

  // ---- issue chunk 0 into buffer 0 ----
#pragma unroll
  for (int r = 0; r < 8; ++r) async_ld(aBase[0] + adst[r], aoff[r], xB);
#pragma unroll
  for (int r = 0; r < 2; ++r) async_ld_nt(bBase[0] + bdst[r], boff[r], wB);

  v8f acc0[4] = {};
  v8f acc1[4] = {};
  const int  mlane = lane & 15;
  const int  ksel  = (lane < 16) ? 0 : 2;   // fragment K-pair select per ISA VGPR layout
  const int  m0    = wave * 32;             // wave owns 32 rows (two 16-row tiles)
  const bool do0   = (row0 + m0)      < cnt;  // wave-uniform: EXEC stays all-ones inside
  const bool do1   = (row0 + m0 + 16) < cnt;

  for (int ko = 0; ko < D; ko += KT) {
    const int cur  = (ko / KT) & 1;
    const bool more = (ko + KT) < D;

    if (more) {                      // issue next chunk into the other buffer (free since last barrier)
      const unsigned kb = (unsigned)((ko + KT) * 4);
#pragma unroll
      for (int r = 0; r < 8; ++r) async_ld(aBase[cur ^ 1] + adst[r], aoff[r] + kb, xB);
#pragma unroll
      for (int r = 0; r < 2; ++r) async_ld_nt(bBase[cur ^ 1] + bdst[r], boff[r] + kb, wB);
      asm volatile("s_wait_asynccnt 10" ::: "memory");   // in-order completion => chunk `cur` resident
    } else {
      asm volatile("s_wait_asynccnt 0" ::: "memory");
    }
    if (ko + 2 * KT < D) {           // keep the weight stream deep
      __builtin_prefetch(We + boff[0] / 4 + ko + 2 * KT, 0, 0);
      __builtin_prefetch(We + boff[1] / 4 + ko + 2 * KT, 0, 0);
    }
    __syncthreads();                 // all waves' async data visible

    if (do0) {
      if (do1) compute_chunk<true >(&As[cur][0], &Bs[cur][0], m0, mlane, ksel, acc0, acc1);
      else     compute_chunk<false>(&As[cur][0], &Bs[cur][0], m0, mlane, ksel, acc0, acc1);
    }
    __syncthreads();                 // buffer `cur` free for overwrite next iteration
  }

  // ---- store C: VGPR v holds M = base + v (lanes 0-15) / base + v + 8 (lanes 16-31) ----
  const int ncol = nt * NT + mlane;
#pragma unroll
  for (int half = 0; half < 2; ++half) {
    const v8f* acc = half ? acc1 : acc0;
    const int mb   = m0 + 16 * half;
    int crow[8];
#pragma unroll
    for (int v = 0; v < 8; ++v) {
      int gr  = row0 + mb + v + ((lane < 16) ? 0 : 8);
      crow[v] = (gr < cnt) ? lists[e * B + gr] : -1;
    }
#pragma unroll
    for (int j = 0; j < 4; ++j) {
#pragma unroll
      for (int v = 0; v < 8; ++v) {
        if (crow[v] >= 0)
          out[(size_t)crow[v] * D + ncol + j * 16] = acc[j][v];
      }
    }
  }
}

extern "C" void kernel_launch(void* const* d_in, const int* in_sizes, int n_in,
                              void* d_out, int out_size, void* d_ws, size_t ws_size,
                              hipStream_t stream) {
  const float* x   = (const float*)d_in[0];
  const float* W   = (const float*)d_in[1];
  const int*   idx = (const int*)d_in[2];
  float*       out = (float*)d_out;

  const int B = in_sizes[2];
  const int D = (int)((long long)in_sizes[0] / B);
  const int E = (int)((long long)in_sizes[1] / ((long long)D * D));

  int* counts = (int*)d_ws;          // 64 ints (first E used)
  int* lists  = counts + 64;         // E * B ints

  zero_i32<<<1, 64, 0, stream>>>(counts, 64);
  zero_f32<<<(out_size + 255) / 256, 256, 0, stream>>>(out, out_size);
  route_kernel<<<(B + 255) / 256, 256, 0, stream>>>(idx, counts, lists, B, E);

  dim3 grid(D / NT, (B + MT - 1) / MT, E);
  moe_wmma_kernel<<<grid, 256, 0, stream>>>(x, W, counts, lists, out, B, D);
}